// PillarFeatureNet_41257455845539
// MI455X (gfx1250) — compile-verified
//
#include <hip/hip_runtime.h>

// ---------------------------------------------------------------------------
// PillarFeatureNet fused pipeline for MI455X (gfx1250, wave32).
//   pass 1: column sums / sumsq of the linear output WITHOUT materializing it
//           (Wcol = col-sums of W, G = W^T W Gram matrix -> per-point
//            contributions v.Wcol and v^T G v).
//   pass 2: per-b (64x9)x(9x64) GEMM via V_WMMA_F32_16X16X4_F32 (K padded
//           9->12, 3 K-steps), fused BN scale/shift + ReLU + max-over-N,
//           LDS transpose (stride 33) -> coalesced 16B-per-lane stores.
// ---------------------------------------------------------------------------

typedef float v2f __attribute__((ext_vector_type(2)));
typedef float v8f __attribute__((ext_vector_type(8)));

#define CIN       9
#define PN        64              // P*N rows per b (= BN feature dim)
#define COUT      64
#define PER_B_IN  (PN * CIN)      // 576 floats
#define PER_B_OUT (COUT * 32)     // 2048 floats (max over N applied)

// workspace layout (floats)
#define WS_WCOL   0     // 9
#define WS_G      16    // 81
#define WS_SUM    128   // 64
#define WS_SQ     192   // 64
#define WS_SCALE  256   // 64
#define WS_SHIFT  320   // 64

// ---------------------------------------------------------------------------
// K1: Wcol, Gram matrix G = W^T W, zero the stat accumulators.
// ---------------------------------------------------------------------------
__global__ void k_prep(const float* __restrict__ W, float* __restrict__ ws) {
    int t = threadIdx.x;                       // 128 threads
    if (t < 81) {
        int r = t / CIN, c = t % CIN;
        float g = 0.f;
        #pragma unroll 8
        for (int o = 0; o < COUT; ++o)
            g = fmaf(W[o * CIN + r], W[o * CIN + c], g);
        ws[WS_G + t] = g;
    } else if (t < 90) {
        int c = t - 81;
        float s = 0.f;
        #pragma unroll 8
        for (int o = 0; o < COUT; ++o) s += W[o * CIN + c];
        ws[WS_WCOL + c] = s;
    }
    ws[WS_SUM + t] = 0.f;                      // zeros SUM[64] + SQ[64]
}

// ---------------------------------------------------------------------------
// K2: per-column (p,n) sums of x and x^2 over all (b, o) rows.
//     sum  += v . Wcol        (v = 9-vector of one point)
//     sumsq+= v^T G v
// ---------------------------------------------------------------------------
__global__ __launch_bounds__(256)
void k_stats(const float* __restrict__ pillars, float* __restrict__ ws,
             int B, int chunk) {
    __shared__ float sW[CIN];
    __shared__ float sG[81];
    __shared__ float sp[4 * PER_B_IN];
    __shared__ float red[512];

    int tid = threadIdx.x;
    if (tid < 81) sG[tid] = ws[WS_G + tid];
    if (tid < CIN) sW[tid] = ws[WS_WCOL + tid];
    __syncthreads();

    int bEnd = min(B, (int)(blockIdx.x + 1) * chunk);
    float a1 = 0.f, a2 = 0.f;

    for (int b0 = blockIdx.x * chunk; b0 < bEnd; b0 += 4) {
        // coalesced stage of 4 b's worth of pillars into LDS
        for (int idx = tid; idx < 4 * PER_B_IN; idx += 256) {
            int bb = b0 + idx / PER_B_IN;
            sp[idx] = (bb < B) ? pillars[(size_t)b0 * PER_B_IN + idx] : 0.f;
        }
        __syncthreads();

        int bsub = tid >> 6, j = tid & 63;
        int b = b0 + bsub;
        if (b < bEnd) {
            const float* vp = &sp[bsub * PER_B_IN + j * CIN];
            float vv[CIN];
            #pragma unroll
            for (int c = 0; c < CIN; ++c) vv[c] = vp[c];
            float c1 = 0.f, c2 = 0.f;
            #pragma unroll
            for (int c = 0; c < CIN; ++c) {
                c1 = fmaf(vv[c], sW[c], c1);
                float t2 = 0.f;
                #pragma unroll
                for (int d = 0; d < CIN; ++d)
                    t2 = fmaf(sG[c * CIN + d], vv[d], t2);
                c2 = fmaf(vv[c], t2, c2);
            }
            a1 += c1; a2 += c2;
        }
        __syncthreads();
    }

    // reduce the 4 b-sub-lanes per column, then one atomic per column
    red[tid] = a1; red[256 + tid] = a2;
    __syncthreads();
    if (tid < 64) {
        float s1 = red[tid] + red[64 + tid] + red[128 + tid] + red[192 + tid];
        float s2 = red[256 + tid] + red[320 + tid] + red[384 + tid] + red[448 + tid];
        atomicAdd(&ws[WS_SUM + tid], s1);
        atomicAdd(&ws[WS_SQ  + tid], s2);
    }
}

// ---------------------------------------------------------------------------
// K3: fold mean/var/gamma/beta into per-column scale & shift.
// ---------------------------------------------------------------------------
__global__ void k_finalize(const float* __restrict__ gamma,
                           const float* __restrict__ beta,
                           float* __restrict__ ws, float invN) {
    int j = threadIdx.x;                       // 64 threads
    float mean = ws[WS_SUM + j] * invN;
    float var  = ws[WS_SQ  + j] * invN - mean * mean;
    float sc   = rsqrtf(var + 1e-5f) * gamma[j];
    ws[WS_SCALE + j] = sc;
    ws[WS_SHIFT + j] = beta[j] - mean * sc;
}

// ---------------------------------------------------------------------------
// K4: main WMMA kernel. 4 waves / block, one b per wave.
//     Per b: 4 row-tiles x 4 col-tiles, each tile = 3x V_WMMA_F32_16X16X4_F32.
// ---------------------------------------------------------------------------
__global__ __launch_bounds__(128)
void k_main(const float* __restrict__ pillars, const float* __restrict__ W,
            const float* __restrict__ ws, float* __restrict__ out, int B) {
    __shared__ float s_in[4][PER_B_IN];        //  9.0 KB
    __shared__ float s_out[4][64 * 33];        // 33.8 KB (stride-33 pad)
    __shared__ float s_scale[64], s_shift[64];

    int tid  = threadIdx.x;
    int w    = tid >> 5;                       // wave id = local b
    int lane = tid & 31;
    int half = lane >> 4;                      // ISA half-wave select
    int l16  = lane & 15;
    int b    = blockIdx.x * 4 + w;

    if (tid < 64)       s_scale[tid]      = ws[WS_SCALE + tid];
    else if (tid < 128) s_shift[tid - 64] = ws[WS_SHIFT + tid - 64];

    // B-operands: Bm[k][o] = W[o][k]; ISA f32 B layout: lanes 0-15 K={k0,k0+1},
    // lanes 16-31 K={k0+2,k0+3}, N = lane%16. Zero-pad K 9..11.
    v2f bm[3][4];
    #pragma unroll
    for (int ks = 0; ks < 3; ++ks) {
        int kb = ks * 4 + half * 2;
        #pragma unroll
        for (int ct = 0; ct < 4; ++ct) {
            int o = ct * 16 + l16;
            bm[ks][ct].x = (kb     < CIN) ? W[o * CIN + kb    ] : 0.f;
            bm[ks][ct].y = (kb + 1 < CIN) ? W[o * CIN + kb + 1] : 0.f;
        }
    }

    if (b < B) {   // wave-uniform guard: EXEC stays all-ones for WMMA
        const float* src = pillars + (size_t)b * PER_B_IN;
        for (int idx = lane; idx < PER_B_IN; idx += 32)
            s_in[w][idx] = src[idx];
    }
    __syncthreads();

    if (b < B) {
        #pragma unroll
        for (int r = 0; r < 4; ++r) {
            // A-operands for this 16-row strip (same ISA half-wave K layout)
            v2f a[3];
            const float* vrow = &s_in[w][(r * 16 + l16) * CIN];
            #pragma unroll
            for (int ks = 0; ks < 3; ++ks) {
                int kb = ks * 4 + half * 2;
                a[ks].x = (kb     < CIN) ? vrow[kb    ] : 0.f;
                a[ks].y = (kb + 1 < CIN) ? vrow[kb + 1] : 0.f;
            }
            // BN scale/shift for the 16 rows of this strip (row = VGPR+8*half)
            float sc[8], sh[8];
            #pragma unroll
            for (int v = 0; v < 8; ++v) {
                int j = r * 16 + v + half * 8;
                sc[v] = s_scale[j];
                sh[v] = s_shift[j];
            }
            #pragma unroll
            for (int ct = 0; ct < 4; ++ct) {
                v8f acc = {0.f, 0.f, 0.f, 0.f, 0.f, 0.f, 0.f, 0.f};
                #pragma unroll
                for (int ks = 0; ks < 3; ++ks) {
                    acc = __builtin_amdgcn_wmma_f32_16x16x4_f32(
                        /*neg_a=*/false, a[ks], /*neg_b=*/false, bm[ks][ct],
                        /*c_mod=*/(short)0, acc,
                        /*reuse_a=*/false, /*reuse_b=*/false);
                }
                int o = ct * 16 + l16;
                // BN + ReLU + max over N (row pairs 2m,2m+1) -> LDS transpose
                #pragma unroll
                for (int v = 0; v < 8; v += 2) {
                    float z0 = fmaf(acc[v],     sc[v],     sh[v]);
                    float z1 = fmaf(acc[v + 1], sc[v + 1], sh[v + 1]);
                    float m  = fmaxf(fmaxf(z0, z1), 0.f);
                    int p = r * 8 + (v >> 1) + half * 4;
                    s_out[w][o * 33 + p] = m;
                }
            }
        }
    }
    __syncthreads();

    if (b < B) {
        // coalesced, 16B-aligned stores: out[b][o][p], o-major
        float* dst = out + (size_t)b * PER_B_OUT;
        #pragma unroll
        for (int it = 0; it < 16; ++it) {
            int flat = (it * 32 + lane) * 4;
            int o = flat >> 5;
            int p = flat & 31;
            float4 val;
            val.x = s_out[w][o * 33 + p];
            val.y = s_out[w][o * 33 + p + 1];
            val.z = s_out[w][o * 33 + p + 2];
            val.w = s_out[w][o * 33 + p + 3];
            *reinterpret_cast<float4*>(dst + flat) = val;
        }
    }
}

// ---------------------------------------------------------------------------
extern "C" void kernel_launch(void* const* d_in, const int* in_sizes, int n_in,
                              void* d_out, int out_size, void* d_ws, size_t ws_size,
                              hipStream_t stream) {
    const float* pillars = (const float*)d_in[0];
    // d_in[1] = num_points_per_pillar: unused by the reference computation.
    const float* W       = (const float*)d_in[2];
    const float* gamma   = (const float*)d_in[3];
    const float* beta    = (const float*)d_in[4];
    float* out = (float*)d_out;
    float* ws  = (float*)d_ws;

    int B = in_sizes[0] / PER_B_IN;            // 16384 for the reference shapes

    k_prep<<<1, 128, 0, stream>>>(W, ws);

    int statBlocks = 128;
    int chunk = (B + statBlocks - 1) / statBlocks;
    k_stats<<<statBlocks, 256, 0, stream>>>(pillars, ws, B, chunk);

    float invN = 1.0f / ((float)B * (float)COUT);
    k_finalize<<<1, 64, 0, stream>>>(gamma, beta, ws, invN);

    k_main<<<(B + 3) / 4, 128, 0, stream>>>(pillars, W, ws, out, B);
}